// GAT_D_85950885527882
// MI455X (gfx1250) — compile-verified
//
#include <hip/hip_runtime.h>
#include <math.h>

// ---------------------------------------------------------------------------
// GAT pipeline for gfx1250 (MI455X): fp32 WMMA GEMM + L2-resident edge
// softmax/scatter. wave32 throughout.
//
// Roofline: total GEMM work ~8.3 GFLOP (negligible); edge gather/scatter moves
// ~7 GB through L2 (all live tensors < 192 MB L2), so the edge kernels are the
// bottleneck and the GEMM runs in fp32 WMMA (no precision sacrifice needed).
// ---------------------------------------------------------------------------

typedef float v2f __attribute__((ext_vector_type(2)));
typedef float v8f __attribute__((ext_vector_type(8)));

#define N_NODES  100000
#define N_EDGES  1000000
#define E_TOT    (N_EDGES + N_NODES)   // explicit edges + self loops
#define N_LAYERS 15
#define NEG_SLOPE 0.2f
#define SWPITCH  40                    // LDS row pitch: 2*40 % 64 == 16 -> the
                                       // two lane halves (K rows differing by 2)
                                       // hit disjoint 16-bank ranges.

// ---- float atomic max via sign-split ordered-int trick (correct with -inf init)
__device__ __forceinline__ void atomicMaxF(float* addr, float v) {
  if (v >= 0.0f) atomicMax((int*)addr, __float_as_int(v));
  else           atomicMin((unsigned int*)addr, __float_as_uint(v));
}

__device__ __forceinline__ int2 edge_sd(const int* __restrict__ ei, int i) {
  if (i < N_EDGES) return make_int2(ei[i], ei[N_EDGES + i]);
  int v = i - N_EDGES;
  return make_int2(v, v);
}

// ---------------------------------------------------------------------------
// GEMM: C[M,fout] = A[M,K] * W[K,fout]   (K multiple of 4, fout padded to 32)
// 256 threads = 8 waves; wave w -> rows [bx*128 + 16w, +16).
// blockIdx.y -> 32-column slice: two 16-col WMMA tiles per wave sharing one
// A fragment (2 WMMAs per global A load).
// V_WMMA_F32_16X16X4_F32 VGPR layouts (ISA 7.12.2):
//   A 16x4:  lane&15 = M row; v0/v1 hold K = kb, kb+1 with kb = (lane>>4)*2
//   B 4x16:  lane&15 = N col; v0/v1 hold K = kb, kb+1
//   C 16x16: VGPR i: lanes 0-15 -> M=i, lanes 16-31 -> M=i+8; N = lane&15
// Row index is CLAMPED (not predicated) so the hot loop has no exec-mask
// branches; out-of-range lanes compute duplicates that the guarded stores drop.
// ---------------------------------------------------------------------------
__global__ void __launch_bounds__(256)
gemm_wmma(const float* __restrict__ A, const float* __restrict__ W,
          float* __restrict__ C, int M, int K, int fout) {
  __shared__ float sW[128 * SWPITCH];          // K<=128, 32-col slice of W
  const int tid     = threadIdx.x;
  const int colTile = blockIdx.y << 5;
  for (int i = tid; i < K * 32; i += 256) {
    int k = i >> 5, c = i & 31, col = colTile + c;
    sW[k * SWPITCH + c] = (col < fout) ? W[k * fout + col] : 0.0f;
  }
  __syncthreads();

  const int wave = tid >> 5;                   // wave32
  const int lane = tid & 31;
  const int row0 = blockIdx.x * 128 + (wave << 4);
  const int r    = lane & 15;
  const int kb   = (lane >> 4) << 1;           // 0 or 2
  int row = row0 + r;
  if (row > M - 1) row = M - 1;                // clamp -> unconditional loads
  const float* Arow = A + (long long)row * K;

  v8f acc0 = {}, acc1 = {};
  for (int k0 = 0; k0 < K; k0 += 4) {
    v2f a, b0, b1;
    a[0]  = Arow[k0 + kb];
    a[1]  = Arow[k0 + kb + 1];
    b0[0] = sW[(k0 + kb) * SWPITCH + r];
    b0[1] = sW[(k0 + kb + 1) * SWPITCH + r];
    b1[0] = sW[(k0 + kb) * SWPITCH + 16 + r];
    b1[1] = sW[(k0 + kb + 1) * SWPITCH + 16 + r];
    acc0 = __builtin_amdgcn_wmma_f32_16x16x4_f32(false, a, false, b0,
                                                 (short)0, acc0, false, false);
    acc1 = __builtin_amdgcn_wmma_f32_16x16x4_f32(false, a, false, b1,
                                                 (short)0, acc1, false, false);
  }

  const int mb   = (lane >> 4) << 3;           // 0 or 8
  const int col0 = colTile + r;
  const int col1 = colTile + 16 + r;
  if (col0 < fout) {
    for (int i = 0; i < 8; ++i) {
      int m = row0 + mb + i;
      if (m < M) C[(long long)m * fout + col0] = acc0[i];
    }
  }
  if (col1 < fout) {
    for (int i = 0; i < 8; ++i) {
      int m = row0 + mb + i;
      if (m < M) C[(long long)m * fout + col1] = acc1[i];
    }
  }
}

// per-node attention logits + init of segment max/sum
__global__ void __launch_bounds__(256)
node_scores(const float* __restrict__ h, const float* __restrict__ a_src,
            const float* __restrict__ a_dst, float* __restrict__ al_s,
            float* __restrict__ al_d, float* __restrict__ emax,
            float* __restrict__ esum, int fout) {
  int i = blockIdx.x * 256 + threadIdx.x;
  if (i >= N_NODES) return;
  const float* hi = h + (long long)i * fout;
  float s = 0.0f, d = 0.0f;
  for (int f = 0; f < fout; ++f) { float v = hi[f]; s += v * a_src[f]; d += v * a_dst[f]; }
  al_s[i] = s; al_d[i] = d;
  emax[i] = -INFINITY;
  esum[i] = 0.0f;
}

__global__ void __launch_bounds__(256)
zero_buf(float* __restrict__ p, int n) {
  int i = blockIdx.x * 256 + threadIdx.x;
  if (i < n) p[i] = 0.0f;
}

// pass 1: edge logit (leaky relu) + segment max over dst
__global__ void __launch_bounds__(256)
edge_logits(const int* __restrict__ ei, const float* __restrict__ al_s,
            const float* __restrict__ al_d, float* __restrict__ e_buf,
            float* __restrict__ emax) {
  int i = blockIdx.x * 256 + threadIdx.x;
  if (i >= E_TOT) return;
  int2 sd = edge_sd(ei, i);
  float e = al_s[sd.x] + al_d[sd.y];
  e = (e > 0.0f) ? e : NEG_SLOPE * e;
  e_buf[i] = e;
  atomicMaxF(&emax[sd.y], e);
}

// pass 2: exp(e - max) + segment sum over dst
__global__ void __launch_bounds__(256)
edge_exp(const int* __restrict__ ei, const float* __restrict__ e_buf,
         const float* __restrict__ emax, float* __restrict__ ex_buf,
         float* __restrict__ esum) {
  int i = blockIdx.x * 256 + threadIdx.x;
  if (i >= E_TOT) return;
  int2 sd = edge_sd(ei, i);
  float ex = __expf(e_buf[i] - emax[sd.y]);
  ex_buf[i] = ex;
  atomicAdd(&esum[sd.y], ex);
}

// pass 3: alpha = ex / sum  (stored back into e_buf)
__global__ void __launch_bounds__(256)
edge_alpha(const int* __restrict__ ei, const float* __restrict__ ex_buf,
           const float* __restrict__ esum, float* __restrict__ alpha) {
  int i = blockIdx.x * 256 + threadIdx.x;
  if (i >= E_TOT) return;
  int2 sd = edge_sd(ei, i);
  alpha[i] = ex_buf[i] / esum[sd.y];
}

// pass 4: out[dst,:] += alpha * h[src,:]. One thread per (edge, feature);
// consecutive lanes cover consecutive features of the same edge so both the
// h[src] reads and the out[dst] atomics are contiguous bursts (L2-resident).
// fout is always a power of two -> shift/mask indexing, no division.
__global__ void __launch_bounds__(256)
edge_scatter(const int* __restrict__ ei, const float* __restrict__ alpha,
             const float* __restrict__ h, float* __restrict__ out,
             int fout, int shift) {
  unsigned int idx = blockIdx.x * 256u + threadIdx.x;
  int i = (int)(idx >> shift);
  if (i >= E_TOT) return;
  int f = (int)(idx & (unsigned)(fout - 1));
  int2 sd = edge_sd(ei, i);
  float v = alpha[i] * h[(long long)sd.x * fout + f];
  atomicAdd(&out[(long long)sd.y * fout + f], v);
}

// bias + activation (0=none, 1=relu, 2=elu), in place
__global__ void __launch_bounds__(256)
finalize(float* __restrict__ out, const float* __restrict__ bias,
         int total, int fout, int act) {
  int idx = blockIdx.x * 256 + threadIdx.x;
  if (idx >= total) return;
  float v = out[idx] + bias[idx & (fout - 1)];
  if (act == 1)      v = fmaxf(v, 0.0f);
  else if (act == 2) v = (v > 0.0f) ? v : (__expf(v) - 1.0f);
  out[idx] = v;
}

__global__ void __launch_bounds__(256)
copy_out(const float* __restrict__ src, float* __restrict__ dst, int n) {
  int i = blockIdx.x * 256 + threadIdx.x;
  if (i < n) dst[i] = src[i];
}

// ---------------------------------------------------------------------------
extern "C" void kernel_launch(void* const* d_in, const int* in_sizes, int n_in,
                              void* d_out, int out_size, void* d_ws, size_t ws_size,
                              hipStream_t stream) {
  (void)in_sizes; (void)n_in; (void)out_size; (void)ws_size;

  static const int FIN [N_LAYERS] = {64,128,128,64,64,32,32,16,16, 8, 8, 8, 8, 4, 4};
  static const int FOUT[N_LAYERS] = {128,128,64,64,32,32,16,16, 8, 8, 8, 8, 4, 4, 1};
  static const int ACTV[N_LAYERS] = { 1,  1,  1, 1, 1, 1, 1, 1, 2, 2, 1, 1, 1, 1, 0};
  static const int SHFT[N_LAYERS] = { 7,  7,  6, 6, 5, 5, 4, 4, 3, 3, 3, 3, 2, 2, 0};

  const float* x  = (const float*)d_in[0];
  const int*   ei = (const int*)d_in[1];

  // workspace layout (floats)
  float* ws = (float*)d_ws;
  float* hA     = ws;                                  // N x 128: transformed h
  float* hB     = hA + (size_t)N_NODES * 128;          // N x 128: aggregated out
  float* al_s   = hB + (size_t)N_NODES * 128;          // N
  float* al_d   = al_s + N_NODES;                      // N
  float* emax   = al_d + N_NODES;                      // N
  float* esum   = emax + N_NODES;                      // N
  float* e_buf  = esum + N_NODES;                      // E_TOT (logits -> alpha)
  float* ex_buf = e_buf + E_TOT;                       // E_TOT

  const int egrid = (E_TOT + 255) / 256;

  // Buffer plan: h (GEMM output) lives in hA for every layer; the aggregation
  // target lives in hB. hB is also the previous layer's input, but the GEMM
  // (which is the only reader of the input) is stream-ordered before the
  // zeroing of hB, so the aliasing is safe. Layer 0 reads x directly.
  const float* prev = x;
  float* h   = hA;
  float* out = hB;

  for (int l = 0; l < N_LAYERS; ++l) {
    const float* W     = (const float*)d_in[2 + 4 * l + 0];
    const float* a_src = (const float*)d_in[2 + 4 * l + 1];
    const float* a_dst = (const float*)d_in[2 + 4 * l + 2];
    const float* bias  = (const float*)d_in[2 + 4 * l + 3];
    const int fin = FIN[l], fout = FOUT[l];
    const int nf  = N_NODES * fout;

    // 1) h = prev @ W  (fp32 WMMA, 2 N-tiles per wave)
    dim3 ggrid((N_NODES + 127) / 128, (fout + 31) / 32);
    gemm_wmma<<<ggrid, 256, 0, stream>>>(prev, W, h, N_NODES, fin, fout);

    // 2) per-node logits, init segment max/sum
    node_scores<<<(N_NODES + 255) / 256, 256, 0, stream>>>(h, a_src, a_dst,
                                                           al_s, al_d, emax, esum, fout);
    // 3) zero the aggregation target (layer input is dead after the GEMM)
    zero_buf<<<(nf + 255) / 256, 256, 0, stream>>>(out, nf);

    // 4-6) segment softmax over dst
    edge_logits<<<egrid, 256, 0, stream>>>(ei, al_s, al_d, e_buf, emax);
    edge_exp   <<<egrid, 256, 0, stream>>>(ei, e_buf, emax, ex_buf, esum);
    edge_alpha <<<egrid, 256, 0, stream>>>(ei, ex_buf, esum, e_buf);

    // 7) weighted scatter-add of features
    unsigned int total = (unsigned int)E_TOT << SHFT[l];
    edge_scatter<<<(total + 255u) / 256u, 256, 0, stream>>>(ei, e_buf, h, out,
                                                            fout, SHFT[l]);
    // 8) bias + activation
    finalize<<<(nf + 255) / 256, 256, 0, stream>>>(out, bias, nf, fout, ACTV[l]);

    prev = out;   // next layer reads the aggregated result
  }

  // last layer: fout=1, HEADS=1 -> mean over heads is identity. Result in hB.
  copy_out<<<(N_NODES + 255) / 256, 256, 0, stream>>>(prev, (float*)d_out, N_NODES);
}